// BMGCN_9861244912160
// MI455X (gfx1250) — compile-verified
//
#include <hip/hip_runtime.h>
#include <hip/hip_bf16.h>
#include <math.h>

// BMGCN forward for MI455X (gfx1250, wave32).
// N=8192, D=512, C=16, HM=256, HG=64.
// - All GEMM-shaped math through v_wmma_f32_16x16x4_f32 (native f32 WMMA).
// - Adjacency (256MB) streamed with the Tensor Data Mover into LDS
//   (tensor_load_to_lds + s_wait_tensorcnt), adj->g transform done in-place in
//   LDS, WMMA A-operand read from LDS.
// - The NxN propagation matrix g is never materialized in HBM.

#define N_NODES 8192
#define DIM     512
#define NC      16
#define HMID    256
#define HG      64
#define NEGV    (-9.0e15f)

typedef __attribute__((ext_vector_type(2))) float v2f;
typedef __attribute__((ext_vector_type(8))) float v8f;
typedef unsigned int u32x4 __attribute__((ext_vector_type(4)));
typedef int          i32x8 __attribute__((ext_vector_type(8)));
typedef int          i32x4 __attribute__((ext_vector_type(4)));

__device__ __forceinline__ v8f wmma_f32_k4(v2f a, v2f b, v8f c) {
  // D[16x16] += A[16x4] * B[4x16], all f32 (VOP3P V_WMMA_F32_16X16X4_F32)
  return __builtin_amdgcn_wmma_f32_16x16x4_f32(false, a, false, b, (short)0, c,
                                               false, false);
}

__device__ __forceinline__ float dot16(const float* q, const float* __restrict__ b) {
  float s = 0.f;
#pragma unroll
  for (int i = 0; i < 16; ++i) s = fmaf(q[i], b[i], s);
  return s;
}

// ---------------------------------------------------------------------------
// TDM: issue a 2D tile load (tile_d0 x tile_d1 elements of 4B) from a
// row-major tensor (row length stride0) into LDS at lds_byte. Rows are packed
// contiguously in LDS (tile_d0*4 bytes apart). Caller must be wave-uniform.
// D# layout per CDNA5 ISA ch.8: group0 {count, lds_addr, global_addr, type=2},
// group1 {data_size=4B, tensor dims, tile dims, dim0 stride}.
// This toolchain uses the 6-arg builtin form (extra int32x8 group before cpol).
// ---------------------------------------------------------------------------
__device__ __forceinline__ void tdm_load_tile_4B(
    const void* gaddr, unsigned lds_byte, unsigned tile_d0, unsigned tile_d1,
    unsigned tensor_d0, unsigned tensor_d1, unsigned stride0)
{
  unsigned long long ga = (unsigned long long)(uintptr_t)gaddr;
  u32x4 g0;
  g0[0] = 1u;                                            // count=1 (valid D#)
  g0[1] = lds_byte;                                      // lds_addr
  g0[2] = (unsigned)ga;                                  // global_addr[31:0]
  g0[3] = (unsigned)((ga >> 32) & 0x01FFFFFFu) | (2u << 30);  // addr[56:32]|type=2
  i32x8 g1;
  g1[0] = 0x20000;                                       // data_size=2 (4 bytes)
  g1[1] = (int)((tensor_d0 & 0xFFFFu) << 16);            // tensor_dim0[15:0]
  g1[2] = (int)((tensor_d0 >> 16) | ((tensor_d1 & 0xFFFFu) << 16));
  g1[3] = (int)((tensor_d1 >> 16) | (tile_d0 << 16));    // tile_dim0
  g1[4] = (int)(tile_d1 & 0xFFFFu);                      // tile_dim1
  g1[5] = (int)stride0;                                  // tensor_dim0_stride lo
  g1[6] = 0;
  g1[7] = 0;
  i32x4 gz  = {0, 0, 0, 0};
  i32x8 gz8 = {0, 0, 0, 0, 0, 0, 0, 0};
  __builtin_amdgcn_tensor_load_to_lds(g0, g1, gz, gz, gz8, 0);
}

// ---------------------------------------------------------------------------
// Generic f32 WMMA GEMM: D[M,N] = act(A[M,K] @ W[K,N] + bias), row-major.
// One wave per 16x16 tile; K stepped by 4.
// ---------------------------------------------------------------------------
__global__ __launch_bounds__(256) void gemm_wmma_f32(
    const float* __restrict__ A, const float* __restrict__ W,
    const float* __restrict__ bias, float* __restrict__ D,
    int M, int N, int K, int lda, int ldb, int ldd, int relu)
{
  int wave = threadIdx.x >> 5;
  int lane = threadIdx.x & 31;
  int tile = blockIdx.x * 8 + wave;
  int ntiles = N >> 4;
  int total = (M >> 4) * ntiles;
  if (tile >= total) return;  // wave-uniform exit: EXEC all-ones for WMMA
  int mt = tile / ntiles, nt = tile - mt * ntiles;
  int r  = lane & 15;
  int kb = (lane >> 4) << 1;
  const float* arow = A + (size_t)(mt * 16 + r) * lda + kb;
  const float* wcol = W + nt * 16 + r;  // r doubles as B column index
  v8f acc = {};
  for (int k = 0; k < K; k += 4) {
    v2f a = *(const v2f*)(arow + k);
    v2f b;
    b.x = wcol[(size_t)(k + kb) * ldb];
    b.y = wcol[(size_t)(k + kb + 1) * ldb];
    acc = wmma_f32_k4(a, b, acc);
  }
  int r0 = (lane >> 4) << 3;
  int ocol = nt * 16 + (lane & 15);
  float bv = bias ? bias[ocol] : 0.f;
#pragma unroll
  for (int v = 0; v < 8; ++v) {
    float d = acc[v] + bv;
    if (relu) d = fmaxf(d, 0.f);
    D[(size_t)(mt * 16 + r0 + v) * ldd + ocol] = d;
  }
}

// ---------------------------------------------------------------------------
// Split-K WMMA GEMM (for adj@soft_y: K=8192, 256MB A stream needs many waves
// in flight to saturate 23.3 TB/s). Writes per-chunk partials; deterministic
// reduce kernel sums them (no float atomics -> bitwise reproducible).
// ---------------------------------------------------------------------------
__global__ __launch_bounds__(256) void gemm_wmma_f32_splitk(
    const float* __restrict__ A, const float* __restrict__ W,
    float* __restrict__ part, int M, int N, int K, int lda, int ldb, int nchunks)
{
  int wave = threadIdx.x >> 5;
  int lane = threadIdx.x & 31;
  int gidx = blockIdx.x * 8 + wave;
  int ntiles = N >> 4;
  int tiles = (M >> 4) * ntiles;
  if (gidx >= tiles * nchunks) return;
  int chunk = gidx / tiles;
  int tile  = gidx - chunk * tiles;
  int mt = tile / ntiles, nt = tile - mt * ntiles;
  int r  = lane & 15;
  int kb = (lane >> 4) << 1;
  int klen = K / nchunks, kst = chunk * klen;
  const float* arow = A + (size_t)(mt * 16 + r) * lda + kb;
  const float* wcol = W + nt * 16 + r;
  v8f acc = {};
  for (int k = kst; k < kst + klen; k += 4) {
    v2f a = *(const v2f*)(arow + k);
    v2f b;
    b.x = wcol[(size_t)(k + kb) * ldb];
    b.y = wcol[(size_t)(k + kb + 1) * ldb];
    acc = wmma_f32_k4(a, b, acc);
  }
  int r0 = (lane >> 4) << 3;
  int ocol = nt * 16 + (lane & 15);
  float* p = part + (size_t)chunk * M * N;
#pragma unroll
  for (int v = 0; v < 8; ++v)
    p[(size_t)(mt * 16 + r0 + v) * N + ocol] = acc[v];
}

__global__ __launch_bounds__(256) void reduce_splitk(
    const float* __restrict__ part, float* __restrict__ D, int elems, int nchunks)
{
  int i = blockIdx.x * 256 + threadIdx.x;
  if (i >= elems) return;
  float s = 0.f;
  for (int c = 0; c < nchunks; ++c) s += part[(size_t)c * elems + i];
  D[i] = s;
}

// ---------------------------------------------------------------------------
// Row softmax over 16 logits -> B; soft_y = mask ? one_hot : B.
// ---------------------------------------------------------------------------
__global__ __launch_bounds__(256) void softmax_rows16(
    const float* __restrict__ pre, const float* __restrict__ onehot,
    const unsigned char* __restrict__ mask, float* __restrict__ Bm,
    float* __restrict__ softy)
{
  int row = blockIdx.x * blockDim.x + threadIdx.x;
  if (row >= N_NODES) return;
  float v[16];
  const float4* p = (const float4*)(pre + (size_t)row * 16);
#pragma unroll
  for (int i = 0; i < 4; ++i) {
    float4 t = p[i];
    v[4*i] = t.x; v[4*i+1] = t.y; v[4*i+2] = t.z; v[4*i+3] = t.w;
  }
  float mx = v[0];
#pragma unroll
  for (int i = 1; i < 16; ++i) mx = fmaxf(mx, v[i]);
  float s = 0.f;
#pragma unroll
  for (int i = 0; i < 16; ++i) { v[i] = __expf(v[i] - mx); s += v[i]; }
  float inv = 1.f / s;
  bool tm = mask[row] != 0;
#pragma unroll
  for (int i = 0; i < 16; ++i) {
    float b = v[i] * inv;
    Bm[(size_t)row * 16 + i] = b;
    softy[(size_t)row * 16 + i] = tm ? onehot[(size_t)row * 16 + i] : b;
  }
}

// ---------------------------------------------------------------------------
// H = (soft_y^T adj soft_y) / rowsum, Q = rownorm((H H^T)*bias).
// HmT = adj@soft_y [N,16] precomputed. One workgroup, 8 waves split K via WMMA.
// ---------------------------------------------------------------------------
__global__ __launch_bounds__(256) void hq_kernel(
    const float* __restrict__ HmT, const float* __restrict__ softy,
    const float* __restrict__ biasCC, float* __restrict__ Hws,
    float* __restrict__ Qws, float* __restrict__ Hout, float* __restrict__ Qout)
{
  __shared__ float red[8][16][16];
  __shared__ float colp[16][16];
  __shared__ float Hs[16][17];
  __shared__ float Qraw[16][17];
  int tid = threadIdx.x;
  int wave = tid >> 5, lane = tid & 31;
  int r = lane & 15, kb = (lane >> 4) << 1;
  v8f acc = {};
  int k0 = wave * 1024;
  for (int k = k0; k < k0 + 1024; k += 4) {
    v2f a, b;
    a.x = HmT[(size_t)(k + kb) * 16 + r];      // A[r,k] = Hm[r,k] = HmT[k,r]
    a.y = HmT[(size_t)(k + kb + 1) * 16 + r];
    b.x = softy[(size_t)(k + kb) * 16 + r];
    b.y = softy[(size_t)(k + kb + 1) * 16 + r];
    acc = wmma_f32_k4(a, b, acc);
  }
  int r0 = (lane >> 4) << 3;
#pragma unroll
  for (int v = 0; v < 8; ++v) red[wave][r0 + v][lane & 15] = acc[v];
  {  // partial column sums of HmT == rowsum(Hm)
    int c = tid & 15, chunk = tid >> 4;
    float p = 0.f;
    for (int k = chunk * 512; k < chunk * 512 + 512; ++k)
      p += HmT[(size_t)k * 16 + c];
    colp[chunk][c] = p;
  }
  __syncthreads();
  {
    int rr = tid >> 4, cc = tid & 15;
    float h = 0.f;
#pragma unroll
    for (int w = 0; w < 8; ++w) h += red[w][rr][cc];
    float rs = 0.f;
#pragma unroll
    for (int ch = 0; ch < 16; ++ch) rs += colp[ch][rr];
    float hv = h / rs;
    Hs[rr][cc] = hv;
    Hws[rr * 16 + cc]  = hv;
    Hout[rr * 16 + cc] = hv;
  }
  __syncthreads();
  {
    int rr = tid >> 4, cc = tid & 15;
    float q = 0.f;
#pragma unroll
    for (int m = 0; m < 16; ++m) q += Hs[rr][m] * Hs[cc][m];
    Qraw[rr][cc] = q * biasCC[rr * 16 + cc];
  }
  __syncthreads();
  {
    int rr = tid >> 4, cc = tid & 15;
    float rs = 0.f;
#pragma unroll
    for (int m = 0; m < 16; ++m) rs += Qraw[rr][m];
    float qv = Qraw[rr][cc] / rs;
    Qws[rr * 16 + cc]  = qv;
    Qout[rr * 16 + cc] = qv;
  }
}

// ---------------------------------------------------------------------------
// Edge-softmax stats: TDM streams one 32KB adj row into LDS; threads compute
// s_ij = BQ_i . B_j (only where adj>0) in place; block-reduce max and sum(exp).
// ---------------------------------------------------------------------------
__global__ __launch_bounds__(256) void edge_softmax_stats(
    const float* __restrict__ adj, const float* __restrict__ Bm,
    const float* __restrict__ BQ, float* __restrict__ mrow,
    float* __restrict__ isig)
{
  __shared__ float svals[N_NODES];   // adj row, transformed in place to s
  __shared__ float red[256];
  int row = blockIdx.x, tid = threadIdx.x;
  if (tid < 32) {  // wave 0 drives the Tensor Data Mover
    tdm_load_tile_4B(adj + (size_t)row * N_NODES,
                     (unsigned)(uintptr_t)svals,
                     /*tile*/ N_NODES, 1, /*tensor*/ N_NODES, N_NODES, N_NODES);
    __builtin_amdgcn_s_wait_tensorcnt(0);
  }
  float bq[16];
  const float4* q4 = (const float4*)(BQ + (size_t)row * 16);
#pragma unroll
  for (int i = 0; i < 4; ++i) {
    float4 t = q4[i];
    bq[4*i] = t.x; bq[4*i+1] = t.y; bq[4*i+2] = t.z; bq[4*i+3] = t.w;
  }
  __syncthreads();
  for (int j0 = tid * 4; j0 < N_NODES; j0 += 1024) {
    float4 a = *(const float4*)(svals + j0);
    float av[4] = {a.x, a.y, a.z, a.w};
#pragma unroll
    for (int e = 0; e < 4; ++e) {
      int col = j0 + e;
      float s = NEGV;
      if (av[e] > 0.f) s = av[e] * dot16(bq, Bm + (size_t)col * 16);
      svals[col] = s;
    }
  }
  __syncthreads();
  float mx = NEGV;
  for (int j = tid; j < N_NODES; j += 256) mx = fmaxf(mx, svals[j]);
  red[tid] = mx; __syncthreads();
  for (int off = 128; off > 0; off >>= 1) {
    if (tid < off) red[tid] = fmaxf(red[tid], red[tid + off]);
    __syncthreads();
  }
  mx = red[0]; __syncthreads();
  float sm = 0.f;
  for (int j = tid; j < N_NODES; j += 256) sm += __expf(svals[j] - mx);
  red[tid] = sm; __syncthreads();
  for (int off = 128; off > 0; off >>= 1) {
    if (tid < off) red[tid] += red[tid + off];
    __syncthreads();
  }
  if (tid == 0) { mrow[row] = mx; isig[row] = 1.f / red[0]; }
}

#define KCH 512  // adj columns per TDM tile (16 x 512 x 4B = 32KB LDS)

// ---------------------------------------------------------------------------
// Propagation 1: h = relu(g @ XW1 + b1), HG=64. One 16-row band per block.
// Per chunk: TDM loads adj[band16, 512cols] tile -> all 128 threads transform
// adj->g in place in LDS -> each of the 4 waves runs the WMMA chain for its
// own 16-column tile of XW1 (g computed once, consumed 4x, no reduction).
// ---------------------------------------------------------------------------
__global__ __launch_bounds__(128) void gcn_prop1(
    const float* __restrict__ adj, const float* __restrict__ Bm,
    const float* __restrict__ BQ, const float* __restrict__ Xw,
    const float* __restrict__ mrow, const float* __restrict__ isig,
    const float* __restrict__ b1, float* __restrict__ h)
{
  __shared__ float tile[16 * KCH];
  int band = blockIdx.x;
  int wave = threadIdx.x >> 5, lane = threadIdx.x & 31;
  int r = lane & 15, kb = (lane >> 4) << 1;
  // transform-phase assignment: thread owns row trow, cols [tcol0, tcol0+64)
  int trow = threadIdx.x >> 3, tcol0 = (threadIdx.x & 7) * 64;
  float bq[16];
  const float4* q4 = (const float4*)(BQ + (size_t)(band * 16 + trow) * 16);
#pragma unroll
  for (int i = 0; i < 4; ++i) {
    float4 t = q4[i];
    bq[4*i] = t.x; bq[4*i+1] = t.y; bq[4*i+2] = t.z; bq[4*i+3] = t.w;
  }
  float mi = mrow[band * 16 + trow], si = isig[band * 16 + trow];
  v8f acc = {};
  for (int ch = 0; ch < N_NODES / KCH; ++ch) {
    int kbase = ch * KCH;
    if (threadIdx.x < 32) {
      tdm_load_tile_4B(adj + (size_t)(band * 16) * N_NODES + kbase,
                       (unsigned)(uintptr_t)tile,
                       /*tile*/ KCH, 16,
                       /*tensor*/ N_NODES, N_NODES, N_NODES);
      __builtin_amdgcn_s_wait_tensorcnt(0);
    }
    __syncthreads();
    // adj -> g in place
    for (int e = 0; e < 64; e += 4) {
      float4 a = *(const float4*)(tile + trow * KCH + tcol0 + e);
      float av[4] = {a.x, a.y, a.z, a.w};
#pragma unroll
      for (int q = 0; q < 4; ++q) {
        int col = kbase + tcol0 + e + q;
        float g = 0.f;
        if (av[q] > 0.f)
          g = __expf(av[q] * dot16(bq, Bm + (size_t)col * 16) - mi) * si;
        tile[trow * KCH + tcol0 + e + q] = g;
      }
    }
    __syncthreads();
    // WMMA chain: wave w owns XW1 columns [w*16, w*16+16)
    const float* wb = Xw + wave * 16 + (lane & 15);
    for (int k = 0; k < KCH; k += 4) {
      v2f a;
      a.x = tile[r * KCH + k + kb];
      a.y = tile[r * KCH + k + kb + 1];
      v2f b;
      b.x = wb[(size_t)(kbase + k + kb) * HG];
      b.y = wb[(size_t)(kbase + k + kb + 1) * HG];
      acc = wmma_f32_k4(a, b, acc);
    }
    __syncthreads();  // before next chunk's TDM overwrites the tile
  }
  int r0 = (lane >> 4) << 3;
  int ocol = wave * 16 + (lane & 15);
  float bv = b1[ocol];
#pragma unroll
  for (int v = 0; v < 8; ++v) {
    float d = fmaxf(acc[v] + bv, 0.f);
    h[(size_t)(band * 16 + r0 + v) * HG + ocol] = d;
  }
}

// ---------------------------------------------------------------------------
// Propagation 2: output = g @ X2 + b2 (C=16), then row-softmax -> logits.
// Same TDM/LDS pipeline; 4 waves split K within each chunk, LDS-reduce at end.
// ---------------------------------------------------------------------------
__global__ __launch_bounds__(128) void gcn_prop2(
    const float* __restrict__ adj, const float* __restrict__ Bm,
    const float* __restrict__ BQ, const float* __restrict__ X2,
    const float* __restrict__ mrow, const float* __restrict__ isig,
    const float* __restrict__ b2, float* __restrict__ out_logits,
    float* __restrict__ out_output)
{
  __shared__ float tile[16 * KCH];
  __shared__ float red[4][16][16];
  __shared__ float fin[16][17];
  int band = blockIdx.x;
  int wave = threadIdx.x >> 5, lane = threadIdx.x & 31;
  int r = lane & 15, kb = (lane >> 4) << 1;
  int trow = threadIdx.x >> 3, tcol0 = (threadIdx.x & 7) * 64;
  float bq[16];
  const float4* q4 = (const float4*)(BQ + (size_t)(band * 16 + trow) * 16);
#pragma unroll
  for (int i = 0; i < 4; ++i) {
    float4 t = q4[i];
    bq[4*i] = t.x; bq[4*i+1] = t.y; bq[4*i+2] = t.z; bq[4*i+3] = t.w;
  }
  float mi = mrow[band * 16 + trow], si = isig[band * 16 + trow];
  v8f acc = {};
  for (int ch = 0; ch < N_NODES / KCH; ++ch) {
    int kbase = ch * KCH;
    if (threadIdx.x < 32) {
      tdm_load_tile_4B(adj + (size_t)(band * 16) * N_NODES + kbase,
                       (unsigned)(uintptr_t)tile,
                       KCH, 16, N_NODES, N_NODES, N_NODES);
      __builtin_amdgcn_s_wait_tensorcnt(0);
    }
    __syncthreads();
    for (int e = 0; e < 64; e += 4) {
      float4 a = *(const float4*)(tile + trow * KCH + tcol0 + e);
      float av[4] = {a.x, a.y, a.z, a.w};
#pragma unroll
      for (int q = 0; q < 4; ++q) {
        int col = kbase + tcol0 + e + q;
        float g = 0.f;
        if (av[q] > 0.f)
          g = __expf(av[q] * dot16(bq, Bm + (size_t)col * 16) - mi) * si;
        tile[trow * KCH + tcol0 + e + q] = g;
      }
    }
    __syncthreads();
    // waves split the chunk's K range (128 each)
    int ks = wave * (KCH / 4), ke = ks + (KCH / 4);
    const float* xb = X2 + (lane & 15);
    for (int k = ks; k < ke; k += 4) {
      v2f a;
      a.x = tile[r * KCH + k + kb];
      a.y = tile[r * KCH + k + kb + 1];
      v2f b;
      b.x = xb[(size_t)(kbase + k + kb) * 16];
      b.y = xb[(size_t)(kbase + k + kb + 1) * 16];
      acc = wmma_f32_k4(a, b, acc);
    }
    __syncthreads();
  }
  int r0 = (lane >> 4) << 3;
#pragma unroll
  for (int v = 0; v < 8; ++v) red[wave][r0 + v][lane & 15] = acc[v];
  __syncthreads();
  for (int o = threadIdx.x; o < 256; o += 128) {
    int rr = o >> 4, cc = o & 15;
    float s = red[0][rr][cc] + red[1][rr][cc] + red[2][rr][cc] + red[3][rr][cc]
            + b2[cc];
    fin[rr][cc] = s;
    out_output[(size_t)(band * 16 + rr) * 16 + cc] = s;
  }
  __syncthreads();
  if (threadIdx.x < 16) {
    int rr = threadIdx.x;
    float mx = fin[rr][0];
#pragma unroll
    for (int i = 1; i < 16; ++i) mx = fmaxf(mx, fin[rr][i]);
    float e[16], sm = 0.f;
#pragma unroll
    for (int i = 0; i < 16; ++i) { e[i] = __expf(fin[rr][i] - mx); sm += e[i]; }
    float inv = 1.f / sm;
#pragma unroll
    for (int i = 0; i < 16; ++i)
      out_logits[(size_t)(band * 16 + rr) * 16 + i] = e[i] * inv;
  }
}

// ---------------------------------------------------------------------------
// final_loss = -mean(log(logits[idx,label])) - mean(log(B[idx,label]))
// ---------------------------------------------------------------------------
__global__ __launch_bounds__(1024) void loss_kernel(
    const float* __restrict__ logits, const float* __restrict__ Bm,
    const int* __restrict__ idx, const int* __restrict__ label,
    float* __restrict__ loss)
{
  __shared__ float red[1024];
  int t = threadIdx.x;
  int i = idx[t], l = label[t];
  red[t] = __logf(logits[(size_t)i * 16 + l]) + __logf(Bm[(size_t)i * 16 + l]);
  __syncthreads();
  for (int off = 512; off > 0; off >>= 1) {
    if (t < off) red[t] += red[t + off];
    __syncthreads();
  }
  if (t == 0) *loss = -red[0] * (1.0f / 1024.0f);
}

// ---------------------------------------------------------------------------
extern "C" void kernel_launch(void* const* d_in, const int* in_sizes, int n_in,
                              void* d_out, int out_size, void* d_ws,
                              size_t ws_size, hipStream_t stream)
{
  (void)in_sizes; (void)n_in; (void)out_size; (void)ws_size;
  const float* feature = (const float*)d_in[0];
  const float* adj     = (const float*)d_in[1];
  const float* onehot  = (const float*)d_in[2];
  const float* biasCC  = (const float*)d_in[3];
  const float* Wm1     = (const float*)d_in[4];
  const float* bm1     = (const float*)d_in[5];
  const float* Wm2     = (const float*)d_in[6];
  const float* bm2     = (const float*)d_in[7];
  const float* W1      = (const float*)d_in[8];
  const float* b1      = (const float*)d_in[9];
  const float* W2      = (const float*)d_in[10];
  const float* b2      = (const float*)d_in[11];
  const int* idx       = (const int*)d_in[12];
  const int* label     = (const int*)d_in[13];
  const unsigned char* mask = (const unsigned char*)d_in[14];

  float* out      = (float*)d_out;          // tuple, flattened:
  float* logits_o = out;                    // [8192,16]
  float* loss_o   = out + 131072;           // [1]
  float* H_o      = out + 131073;           // [16,16]
  float* Q_o      = out + 131329;           // [16,16]
  float* output_o = out + 131585;           // [8192,16]

  float* ws    = (float*)d_ws;              // ~24.2 MB total
  float* Z1    = ws;                        // 8192*256
  float* Bpre  = Z1    + 2097152;           // 8192*16
  float* Bm    = Bpre  + 131072;
  float* softy = Bm    + 131072;
  float* HmT   = softy + 131072;            // 8192*16  (= adj @ soft_y = Hm^T)
  float* Hws   = HmT   + 131072;            // 256
  float* Qws   = Hws   + 256;               // 256
  float* BQ    = Qws   + 256;               // 8192*16
  float* XW1   = BQ    + 131072;            // 8192*64
  float* hbuf  = XW1   + 524288;            // 8192*64
  float* X2    = hbuf  + 524288;            // 8192*16
  float* mrow  = X2    + 131072;            // 8192
  float* isigw = mrow  + 8192;              // 8192
  float* partk = isigw + 8192;              // 16 * 8192*16 split-K partials

  // 1) Z1 = relu(feature @ Wm1 + bm1)            [8192,256]
  gemm_wmma_f32<<<dim3(1024), dim3(256), 0, stream>>>(
      feature, Wm1, bm1, Z1, 8192, 256, 512, 512, 256, 256, 1);
  // 2) Bpre = Z1 @ Wm2 + bm2                     [8192,16]
  gemm_wmma_f32<<<dim3(64), dim3(256), 0, stream>>>(
      Z1, Wm2, bm2, Bpre, 8192, 16, 256, 256, 16, 16, 0);
  // 3) XW1 = feature @ W1                        [8192,64]
  gemm_wmma_f32<<<dim3(256), dim3(256), 0, stream>>>(
      feature, W1, nullptr, XW1, 8192, 64, 512, 512, 64, 64, 0);
  // 4) B = softmax(Bpre); soft_y = mask ? onehot : B
  softmax_rows16<<<dim3(32), dim3(256), 0, stream>>>(Bpre, onehot, mask, Bm, softy);
  // 5) HmT = adj @ soft_y, split-K x16 for HBM saturation, then reduce
  gemm_wmma_f32_splitk<<<dim3(1024), dim3(256), 0, stream>>>(
      adj, softy, partk, 8192, 16, 8192, 8192, 16, 16);
  reduce_splitk<<<dim3(512), dim3(256), 0, stream>>>(partk, HmT, 131072, 16);
  // 6) H, Q (also written to d_out)
  hq_kernel<<<dim3(1), dim3(256), 0, stream>>>(HmT, softy, biasCC, Hws, Qws, H_o, Q_o);
  // 7) BQ = B @ Q                                [8192,16]
  gemm_wmma_f32<<<dim3(64), dim3(256), 0, stream>>>(
      Bm, Qws, nullptr, BQ, 8192, 16, 16, 16, 16, 16, 0);
  // 8) per-row masked-softmax stats over adj (TDM-staged rows)
  edge_softmax_stats<<<dim3(8192), dim3(256), 0, stream>>>(adj, Bm, BQ, mrow, isigw);
  // 9) h = relu(g @ XW1 + b1)                    [8192,64]
  gcn_prop1<<<dim3(512), dim3(128), 0, stream>>>(
      adj, Bm, BQ, XW1, mrow, isigw, b1, hbuf);
  // 10) X2 = h @ W2                              [8192,16]
  gemm_wmma_f32<<<dim3(64), dim3(256), 0, stream>>>(
      hbuf, W2, nullptr, X2, 8192, 16, 64, 64, 16, 16, 0);
  // 11) output = g @ X2 + b2; logits = softmax(output)
  gcn_prop2<<<dim3(512), dim3(128), 0, stream>>>(
      adj, Bm, BQ, X2, mrow, isigw, b2, logits_o, output_o);
  // 12) final loss
  loss_kernel<<<dim3(1), dim3(1024), 0, stream>>>(logits_o, Bm, idx, label, loss_o);
}